// OUR_CUSTOM_SSIM_PYRAMID_7121055777038
// MI455X (gfx1250) — compile-verified
//
#include <hip/hip_runtime.h>
#include <hip/hip_bf16.h>
#include <math.h>

// ---------------------------------------------------------------------------
// SSIM-pyramid loss for (2,1,512,512) f32 inputs on gfx1250 (MI455X).
// Window sums are computed as banded 0/1 matmuls on V_WMMA_F32_16X16X4_F32.
// ---------------------------------------------------------------------------

typedef float v2f __attribute__((ext_vector_type(2)));
typedef float v8f __attribute__((ext_vector_type(8)));

#define EPS2 1e-10f
#define LW   28     // padded halo width (26 valid + 2 pad for K%4==0)

static __device__ __forceinline__ v8f wmma4(v2f a, v2f b, v8f c) {
    // 8 args: (neg_a, A, neg_b, B, c_mod, C, reuse_a, reuse_b)
    return __builtin_amdgcn_wmma_f32_16x16x4_f32(
        false, a, false, b, (short)0, c, false, false);
}

static __device__ __forceinline__ float mval(const float* t1, const float* t2,
                                             int idx, int mi) {
    float a = t1[idx], b = t2[idx];
    switch (mi) {
        case 0:  return a;
        case 1:  return b;
        case 2:  return a * a;
        case 3:  return b * b;
        default: return a * b;
    }
}

// One wave (32 threads) per 16x16 output tile per batch image.
// Computes per-pixel MSE contribution of one pyramid level, writes one
// partial sum per tile (deterministic; no float atomics).
__global__ __launch_bounds__(32)
void OUR_CUSTOM_SSIM_PYRAMID_stats_kernel(const float* __restrict__ i1,
                                          const float* __restrict__ i2,
                                          int H, int W,
                                          float* __restrict__ partials) {
    __shared__ float t1[32 * LW];
    __shared__ float t2[32 * LW];
    __shared__ float Hb[32 * 16];

    const int l  = threadIdx.x;          // lane 0..31
    const int tx = blockIdx.x, ty = blockIdx.y, b = blockIdx.z;
    const float* p1 = i1 + (size_t)b * H * W;
    const float* p2 = i2 + (size_t)b * H * W;

    // Cooperative zero-padded halo load: rows/cols [-5, +20] of the tile.
    // 32*28 = 896 = 32 lanes * 28 iters -> fully uniform (EXEC stays all-1s).
    for (int idx = l; idx < 32 * LW; idx += 32) {
        int r = idx / LW, c = idx % LW;
        int gr = ty * 16 + r - 5, gc = tx * 16 + c - 5;
        bool ok = (r < 26) && (c < 26) && (gr >= 0) && (gr < H) &&
                  (gc >= 0) && (gc < W);
        t1[idx] = ok ? p1[gr * W + gc] : 0.0f;
        t2[idx] = ok ? p2[gr * W + gc] : 0.0f;
    }
    __syncthreads();

    const int lo = l & 15;       // M / N index
    const int hi = l >> 4;       // half-wave select
    v8f S[5];

    #pragma unroll
    for (int mi = 0; mi < 5; ++mi) {
        // ---- horizontal band matmul: H(32x16) = X(32x28) * S^T(28x16) ----
        #pragma unroll
        for (int rb = 0; rb < 2; ++rb) {
            v8f hacc = {};
            #pragma unroll
            for (int kc = 0; kc < 7; ++kc) {
                int row = rb * 16 + lo;
                int c0  = 4 * kc + 2 * hi;                 // A: K = 2*hi + j
                v2f a;
                a.x = mval(t1, t2, row * LW + c0,     mi);
                a.y = mval(t1, t2, row * LW + c0 + 1, mi);
                v2f bb;                                    // B: K = 2*hi + j
                bb.x = (c0     >= lo && c0     <= lo + 10) ? 1.0f : 0.0f;
                bb.y = (c0 + 1 >= lo && c0 + 1 <= lo + 10) ? 1.0f : 0.0f;
                hacc = wmma4(a, bb, hacc);
            }
            #pragma unroll
            for (int v = 0; v < 8; ++v) {                  // D: M = v + 8*hi
                int row = rb * 16 + v + 8 * hi;
                Hb[row * 16 + lo] = hacc[v];
            }
        }
        __syncthreads();
        // ---- vertical band matmul: V(16x16) = S(16x28) * H(28x16) ----
        v8f vacc = {};
        #pragma unroll
        for (int kc = 0; kc < 7; ++kc) {
            int k0 = 4 * kc + 2 * hi;
            v2f a;
            a.x = (k0     >= lo && k0     <= lo + 10) ? 1.0f : 0.0f;
            a.y = (k0 + 1 >= lo && k0 + 1 <= lo + 10) ? 1.0f : 0.0f;
            v2f bb;
            bb.x = Hb[k0 * 16 + lo];
            bb.y = Hb[(k0 + 1) * 16 + lo];
            vacc = wmma4(a, bb, vacc);
        }
        S[mi] = vacc;
        __syncthreads();
    }

    // ---- per-pixel contribution from the five window sums ----
    const float inv121 = 1.0f / 121.0f;
    float lanesum = 0.0f;
    #pragma unroll
    for (int v = 0; v < 8; ++v) {
        float S1 = S[0][v], S2 = S[1][v];
        float S11 = S[2][v], S22 = S[3][v], S12 = S[4][v];
        float mu1 = S1 * inv121, mu2 = S2 * inv121;
        float sg1 = fmaxf(S11 * inv121 - mu1 * mu1, 0.0f) + EPS2;
        float sg2 = fmaxf(S22 * inv121 - mu2 * mu2, 0.0f) + EPS2;
        float ia = 1.0f / (sqrtf(sg1) + EPS2);
        float ib = 1.0f / (sqrtf(sg2) + EPS2);
        float c  = ia * mu1 - ib * mu2;
        float term = ia * ia * S11 + ib * ib * S22 + 121.0f * c * c
                   - 2.0f * ia * ib * S12 - 2.0f * ia * c * S1
                   + 2.0f * ib * c * S2;
        lanesum += term;
    }
    // wave32 reduction (deterministic tree)
    #pragma unroll
    for (int off = 16; off > 0; off >>= 1)
        lanesum += __shfl_down(lanesum, off, 32);
    if (l == 0) {
        int tileIdx = (b * gridDim.y + ty) * gridDim.x + tx;
        partials[tileIdx] = lanesum;
    }
}

// F.interpolate(scale=0.5, bicubic a=-0.75, align_corners=False):
// fixed separable 4-tap stride-2 filter with edge-clamped borders.
__global__ __launch_bounds__(256)
void OUR_CUSTOM_SSIM_PYRAMID_down_kernel(const float* __restrict__ in,
                                         float* __restrict__ out,
                                         int H, int W) {
    const int Ho = H >> 1, Wo = W >> 1;
    const int total = 2 * Ho * Wo;           // B = 2
    int idx = blockIdx.x * blockDim.x + threadIdx.x;
    if (idx >= total) return;
    int x = idx % Wo;
    int t = idx / Wo;
    int y = t % Ho;
    int b = t / Ho;
    const float* p = in + (size_t)b * H * W;
    const float k[4] = {-0.09375f, 0.59375f, 0.59375f, -0.09375f};
    float acc = 0.0f;
    #pragma unroll
    for (int i = 0; i < 4; ++i) {
        int yy = min(max(2 * y - 1 + i, 0), H - 1);
        float rowv = 0.0f;
        #pragma unroll
        for (int j = 0; j < 4; ++j) {
            int xx = min(max(2 * x - 1 + j, 0), W - 1);
            rowv += k[j] * p[yy * W + xx];
        }
        acc += k[i] * rowv;
    }
    out[idx] = acc;
}

// Deterministic fixed-order reduction of all level partials -> scalar loss.
__global__ __launch_bounds__(256)
void OUR_CUSTOM_SSIM_PYRAMID_final_kernel(const float* __restrict__ partials,
                                          float* __restrict__ out) {
    __shared__ float red[256];
    const int t = threadIdx.x;
    const int   offs[4] = {0, 2048, 2560, 2688};
    const int   cnts[4] = {2048, 512, 128, 32};
    const float wgt[4]  = {0.2f, 0.4f, 0.6f, 0.8f};
    const int   dims[4] = {512, 256, 128, 64};
    float total = 0.0f;
    for (int lv = 0; lv < 4; ++lv) {
        float s = 0.0f;
        for (int i = t; i < cnts[lv]; i += 256) s += partials[offs[lv] + i];
        red[t] = s;
        __syncthreads();
        for (int st = 128; st > 0; st >>= 1) {
            if (t < st) red[t] += red[t + st];
            __syncthreads();
        }
        if (t == 0) {
            float denom = 2.0f * 121.0f * (float)dims[lv] * (float)dims[lv];
            total += wgt[lv] * (red[0] / denom);
        }
        __syncthreads();
    }
    if (t == 0) out[0] = total;
}

extern "C" void kernel_launch(void* const* d_in, const int* in_sizes, int n_in,
                              void* d_out, int out_size, void* d_ws, size_t ws_size,
                              hipStream_t stream) {
    const float* img1 = (const float*)d_in[0];
    const float* img2 = (const float*)d_in[1];
    float* out = (float*)d_out;
    float* ws  = (float*)d_ws;

    // workspace layout (floats)
    float* ds1a  = ws;               // 2*256*256 = 131072
    float* ds1b  = ws + 131072;
    float* ds2a  = ws + 262144;      // 2*128*128 = 32768
    float* ds2b  = ws + 294912;
    float* ds3a  = ws + 327680;      // 2*64*64 = 8192
    float* ds3b  = ws + 335872;
    float* parts = ws + 344064;      // 2048 + 512 + 128 + 32 partials

    dim3 wave(32, 1, 1);

    // level 0: 512x512
    OUR_CUSTOM_SSIM_PYRAMID_stats_kernel<<<dim3(32, 32, 2), wave, 0, stream>>>(
        img1, img2, 512, 512, parts + 0);
    int n1 = 2 * 256 * 256;
    OUR_CUSTOM_SSIM_PYRAMID_down_kernel<<<(n1 + 255) / 256, 256, 0, stream>>>(
        img1, ds1a, 512, 512);
    OUR_CUSTOM_SSIM_PYRAMID_down_kernel<<<(n1 + 255) / 256, 256, 0, stream>>>(
        img2, ds1b, 512, 512);

    // level 1: 256x256
    OUR_CUSTOM_SSIM_PYRAMID_stats_kernel<<<dim3(16, 16, 2), wave, 0, stream>>>(
        ds1a, ds1b, 256, 256, parts + 2048);
    int n2 = 2 * 128 * 128;
    OUR_CUSTOM_SSIM_PYRAMID_down_kernel<<<(n2 + 255) / 256, 256, 0, stream>>>(
        ds1a, ds2a, 256, 256);
    OUR_CUSTOM_SSIM_PYRAMID_down_kernel<<<(n2 + 255) / 256, 256, 0, stream>>>(
        ds1b, ds2b, 256, 256);

    // level 2: 128x128
    OUR_CUSTOM_SSIM_PYRAMID_stats_kernel<<<dim3(8, 8, 2), wave, 0, stream>>>(
        ds2a, ds2b, 128, 128, parts + 2560);
    int n3 = 2 * 64 * 64;
    OUR_CUSTOM_SSIM_PYRAMID_down_kernel<<<(n3 + 255) / 256, 256, 0, stream>>>(
        ds2a, ds3a, 128, 128);
    OUR_CUSTOM_SSIM_PYRAMID_down_kernel<<<(n3 + 255) / 256, 256, 0, stream>>>(
        ds2b, ds3b, 128, 128);

    // level 3: 64x64
    OUR_CUSTOM_SSIM_PYRAMID_stats_kernel<<<dim3(4, 4, 2), wave, 0, stream>>>(
        ds3a, ds3b, 64, 64, parts + 2688);

    OUR_CUSTOM_SSIM_PYRAMID_final_kernel<<<1, 256, 0, stream>>>(parts, out);
}